// DenoiserPairFeatures_37967510897122
// MI455X (gfx1250) — compile-verified
//
#include <hip/hip_runtime.h>
#include <hip/hip_bf16.h>

#define AS1 __attribute__((address_space(1)))
#define AS3 __attribute__((address_space(3)))

typedef int   v4i_t __attribute__((vector_size(16)));     // builtin's int4 param type
typedef float v4f   __attribute__((ext_vector_type(4)));  // packed-f32 friendly math

#if defined(__has_builtin)
#if __has_builtin(__builtin_amdgcn_global_load_async_to_lds_b128)
#define HAVE_ASYNC_LDS 1
#endif
#endif

namespace {
constexpr int N       = 1024;
constexpr int C_IN    = 219;
constexpr int C_OUT   = 256;
constexpr int USED    = 187;          // used W columns: sep(127) + xt(30) + xsc(30)
constexpr int TROWS   = USED + 3;     // + bias, gamma, beta rows
constexpr int TFLOATS = TROWS * C_OUT;          // 48640 floats = 194,560 B (fits 320KB WGP LDS)
constexpr int BIAS_ROW  = USED;
constexpr int GAMMA_ROW = USED + 1;
constexpr int BETA_ROW  = USED + 2;

// workspace layout: [table: 194,560 B][pad][desc: N*N*4 = 4 MiB]
constexpr size_t DESC_OFF_BYTES = 196608;   // 192 KiB-aligned region after table

constexpr int THREADS         = 1024;  // 32 wave32s per workgroup
constexpr int WAVES_PER_BLOCK = THREADS / 32;
constexpr int PAIRS_PER_WAVE  = 4;
constexpr int PAIRS_PER_BLOCK = WAVES_PER_BLOCK * PAIRS_PER_WAVE;  // 128
constexpr int GRID            = (N * N) / PAIRS_PER_BLOCK;         // 8192
}

// ---------------------------------------------------------------------------
// Prep 1: pack W^T (used columns only) + b + gamma + beta into a contiguous
// [190][256] fp32 table: each one-hot class is a contiguous 1 KiB row.
// ---------------------------------------------------------------------------
__global__ void pack_table_kernel(const float* __restrict__ W,
                                  const float* __restrict__ b,
                                  const float* __restrict__ gamma,
                                  const float* __restrict__ beta,
                                  float* __restrict__ table) {
  int idx = blockIdx.x * blockDim.x + threadIdx.x;
  if (idx >= TFLOATS) return;
  int r = idx >> 8;        // table row (W column index / aux row)
  int c = idx & 255;       // output channel
  float v;
  if (r < USED)            v = W[c * C_IN + r];
  else if (r == BIAS_ROW)  v = b[c];
  else if (r == GAMMA_ROW) v = gamma[c];
  else                     v = beta[c];
  table[idx] = v;
}

// searchsorted(linspace(0.1, 3.0, 29), d): count of edges strictly < d
__device__ __forceinline__ int dist_bin(const float* __restrict__ X, int i, int j) {
  float dx = X[3 * i + 0] - X[3 * j + 0];
  float dy = X[3 * i + 1] - X[3 * j + 1];
  float dz = X[3 * i + 2] - X[3 * j + 2];
  float d  = sqrtf(dx * dx + dy * dy + dz * dz + 1e-10f);
  int idx = 0;
#pragma unroll
  for (int k = 0; k < 29; ++k) {
    float e = 0.1f + (2.9f / 28.0f) * (float)k;   // folded to constants
    idx += (e < d) ? 1 : 0;
  }
  return idx;
}

// ---------------------------------------------------------------------------
// Prep 2: one THREAD per pair computes all one-hot indices once (vs. 32-lane
// redundant per-wave evaluation in the main kernel) and packs them:
//   desc = sidx | (t_bin << 8) | (sc_bin << 16), or -1 if the pair is masked.
// ---------------------------------------------------------------------------
__global__ void bin_kernel(const int*   __restrict__ mask,
                           const float* __restrict__ x_t,
                           const float* __restrict__ x_sc,
                           int*         __restrict__ desc) {
  const int pair = blockIdx.x * blockDim.x + threadIdx.x;   // grid covers N*N
  const int i = pair >> 10;
  const int j = pair & (N - 1);
  int d = -1;
  if (mask[i] & mask[j]) {
    const int sidx = min(max(i - j + 63, 0), 126);          // seq-sep searchsorted
    const int tb   = dist_bin(x_t,  i, j);
    const int sb   = dist_bin(x_sc, i, j);
    d = sidx | (tb << 8) | (sb << 16);
  }
  desc[pair] = d;
}

// ---------------------------------------------------------------------------
// Main: one wave32 per (i,j) pair.
//   y = Wt[sep] + Wt[127+t] + Wt[157+sc] + b ; LayerNorm(256) ; masked -> 0
// Lane owns channels [4L,4L+4) and [128+4L,128+4L+4): b128 LDS gathers,
// packed-f32 math, wave32 shuffle reduction, b128 NT stores.
// ---------------------------------------------------------------------------
__global__ void __launch_bounds__(THREADS)
pair_kernel(const int*   __restrict__ desc,
            const float* __restrict__ table,
            float*       __restrict__ out) {
  extern __shared__ float lds[];   // TFLOATS floats (194.5 KB — CDNA5 WGP LDS)

  const int tid = threadIdx.x;

  // ---- Stage the whole table into LDS (gfx1250 async-to-LDS path) ----
  {
    const float4* src4 = (const float4*)table;
#ifdef HAVE_ASYNC_LDS
    for (int v = tid; v < TFLOATS / 4; v += THREADS) {
      __builtin_amdgcn_global_load_async_to_lds_b128(
          (AS1 v4i_t*)(src4 + v),
          (AS3 v4i_t*)((float4*)lds + v),
          /*offset=*/0, /*cpol=*/0);
    }
#if __has_builtin(__builtin_amdgcn_s_wait_asynccnt)
    __builtin_amdgcn_s_wait_asynccnt(0);
#else
    asm volatile("s_wait_asynccnt 0" ::: "memory");
#endif
#else
    for (int v = tid; v < TFLOATS / 4; v += THREADS) {
      ((float4*)lds)[v] = src4[v];
    }
#endif
  }
  __syncthreads();

  const int lane = tid & 31;
  const int wave = tid >> 5;
  const v4f* __restrict__ lds4 = (const v4f*)lds;   // 64 v4f per table row

  // Pair-invariant rows: hoisted out of the pair loop (kept in VGPRs).
  const v4f bias0 = lds4[BIAS_ROW  * 64 + lane];
  const v4f bias1 = lds4[BIAS_ROW  * 64 + 32 + lane];
  const v4f gam0  = lds4[GAMMA_ROW * 64 + lane];
  const v4f gam1  = lds4[GAMMA_ROW * 64 + 32 + lane];
  const v4f bet0  = lds4[BETA_ROW  * 64 + lane];
  const v4f bet1  = lds4[BETA_ROW  * 64 + 32 + lane];

#pragma unroll 1
  for (int it = 0; it < PAIRS_PER_WAVE; ++it) {
    // wave-uniform pair index -> scalar descriptor load
    const int pair = __builtin_amdgcn_readfirstlane(
        (blockIdx.x * WAVES_PER_BLOCK + wave) * PAIRS_PER_WAVE + it);
    const int d = desc[pair];
    v4f* __restrict__ po4 = (v4f*)(out + ((size_t)pair << 8));

    if (d < 0) {
      // masked pair: y = LN(b)*gamma+beta then * 0  ==> exactly zero
      const v4f z = {0.0f, 0.0f, 0.0f, 0.0f};
      __builtin_nontemporal_store(z, po4 + lane);
      __builtin_nontemporal_store(z, po4 + 32 + lane);
      continue;
    }

    const int sidx = d & 0xff;
    const int tb   = (d >> 8) & 0xff;
    const int sb   = (d >> 16) & 0xff;

    const v4f* __restrict__ rowS = lds4 + sidx * 64;
    const v4f* __restrict__ rowT = lds4 + (127 + tb) * 64;
    const v4f* __restrict__ rowC = lds4 + (157 + sb) * 64;

    // 8 channels per lane as two b128 gathers per row (conflict-free)
    const v4f y0 = rowS[lane]      + rowT[lane]      + rowC[lane]      + bias0;
    const v4f y1 = rowS[32 + lane] + rowT[32 + lane] + rowC[32 + lane] + bias1;

    float s0 = (y0.x + y0.y) + (y0.z + y0.w) + (y1.x + y1.y) + (y1.z + y1.w);
    v4f  sq  = y0 * y0 + y1 * y1;
    float s1 = (sq.x + sq.y) + (sq.z + sq.w);

    // wave32 butterfly reduction for sum and sum-of-squares
#pragma unroll
    for (int off = 16; off > 0; off >>= 1) {
      s0 += __shfl_xor(s0, off, 32);
      s1 += __shfl_xor(s1, off, 32);
    }

    const float mu   = s0 * (1.0f / 256.0f);
    const float var  = s1 * (1.0f / 256.0f) - mu * mu;
    const float rinv = rsqrtf(var + 1e-5f);

    const v4f muv = {mu, mu, mu, mu};
    const v4f rv  = {rinv, rinv, rinv, rinv};
    const v4f o0  = (y0 - muv) * rv * gam0 + bet0;
    const v4f o1  = (y1 - muv) * rv * gam1 + bet1;

    // 1 GiB output stream: non-temporal b128 stores, 512 B/wave/instruction
    __builtin_nontemporal_store(o0, po4 + lane);
    __builtin_nontemporal_store(o1, po4 + 32 + lane);
  }
}

// ---------------------------------------------------------------------------
extern "C" void kernel_launch(void* const* d_in, const int* in_sizes, int n_in,
                              void* d_out, int out_size, void* d_ws, size_t ws_size,
                              hipStream_t stream) {
  const int*   mask  = (const int*)d_in[0];
  const float* x_t   = (const float*)d_in[1];
  const float* x_sc  = (const float*)d_in[2];
  const float* W     = (const float*)d_in[3];
  const float* b     = (const float*)d_in[4];
  const float* gamma = (const float*)d_in[5];
  const float* beta  = (const float*)d_in[6];
  float*       out   = (float*)d_out;

  float* table = (float*)d_ws;                              // 194,560 B
  int*   desc  = (int*)((char*)d_ws + DESC_OFF_BYTES);      // 4 MiB

  pack_table_kernel<<<(TFLOATS + 255) / 256, 256, 0, stream>>>(W, b, gamma, beta, table);
  bin_kernel<<<(N * N) / 256, 256, 0, stream>>>(mask, x_t, x_sc, desc);

  const size_t ldsBytes = (size_t)TFLOATS * sizeof(float);
  (void)hipFuncSetAttribute((const void*)pair_kernel,
                            hipFuncAttributeMaxDynamicSharedMemorySize,
                            (int)ldsBytes);
  pair_kernel<<<GRID, THREADS, ldsBytes, stream>>>(desc, table, out);
}